// SOM_26036091748934
// MI455X (gfx1250) — compile-verified
//
#include <hip/hip_runtime.h>

// SOM forward on MI455X (gfx1250): fp32 WMMA GEMM (v_wmma_f32_16x16x4_f32),
// full K-row of distances held in the 320KB WGP LDS, fused min/argmin + output.

typedef float v2f __attribute__((ext_vector_type(2)));
typedef float v8f __attribute__((ext_vector_type(8)));

#define SOM_N 16384
#define SOM_D 1024
#define SOM_K 4096
#define SOM_EPS 1e-6f
#define SOM_INV_SIG2 (1.0f / 1024.0f)   // 1/SIGMA^2, SIGMA=32

// ---------------------------------------------------------------- row stats
// x2[n] = sum_d x[n,d]^2 ; sx[n] = sum_d x[n,d].  One wave32 per row.
__global__ __launch_bounds__(256) void som_row_stats(
    const float* __restrict__ x, float* __restrict__ x2, float* __restrict__ sx) {
  int row  = (blockIdx.x * blockDim.x + threadIdx.x) >> 5;
  int lane = threadIdx.x & 31;
  if (row >= SOM_N) return;
  const float* r = x + (size_t)row * SOM_D;
  float s2 = 0.f, s1 = 0.f;
  for (int i = lane; i < SOM_D; i += 32) {
    float v = r[i];
    s2 += v * v;
    s1 += v;
  }
  for (int off = 16; off > 0; off >>= 1) {
    s2 += __shfl_down(s2, off, 32);
    s1 += __shfl_down(s1, off, 32);
  }
  if (lane == 0) { x2[row] = s2; sx[row] = s1; }
}

// ---------------------------------------------------------------- col stats
// w2[k] = sum_d weight[d,k]^2 ; sw[k] = sum_d weight[d,k]. Thread per column.
__global__ __launch_bounds__(256) void som_col_stats(
    const float* __restrict__ w, float* __restrict__ w2, float* __restrict__ sw) {
  int k = blockIdx.x * blockDim.x + threadIdx.x;
  if (k >= SOM_K) return;
  float s2 = 0.f, s1 = 0.f;
  for (int d = 0; d < SOM_D; ++d) {
    float v = w[(size_t)d * SOM_K + k];   // consecutive lanes -> coalesced
    s2 += v * v;
    s1 += v;
  }
  w2[k] = s2;
  sw[k] = s1;
}

__global__ void som_zero(float* p) { if (threadIdx.x == 0) *p = 0.f; }

// ------------------------------------------------------------------- main
// One workgroup = 16-row tile. 8 waves, each wave owns 32 of the 256
// 16x16 N-tiles. Full dist row set (16 x 4096 fp32 = 256KB) lives in LDS.
__global__ __launch_bounds__(256) void som_main(
    const float* __restrict__ x, const float* __restrict__ w,
    const float* __restrict__ x2g, const float* __restrict__ sxg,
    const float* __restrict__ w2g, const float* __restrict__ swg,
    float* __restrict__ bmu_out, float* __restrict__ loss_out,
    float* __restrict__ som_out) {
  __shared__ float s_dist[16 * SOM_K];   // 256 KB
  __shared__ float s_x2[16], s_sx[16];
  __shared__ float s_pmin[256];
  __shared__ int   s_pidx[256];
  __shared__ float s_rowmin[16], s_bi[16], s_bj[16];

  const int tid  = threadIdx.x;
  const int wave = tid >> 5;
  const int lane = tid & 31;
  const int row0 = blockIdx.x << 4;

  if (tid < 16) {
    s_x2[tid] = x2g[row0 + tid];
    s_sx[tid] = sxg[row0 + tid];
  }
  __syncthreads();

  // ---- Stage 1: distances via WMMA f32 16x16x4 ----
  // A (16x4 fp32): lane l holds row m=l&15; VGPR j holds K = 2*(l>>4) + j.
  // B ( 4x16 fp32): lane l holds col n=l&15; VGPR j holds K = 2*(l>>4) + j.
  // C/D (16x16):    lane l, VGPR r -> (m = r + 8*(l>>4), n = l&15).
  const int mh = (lane >> 4);            // 0 or 1 (lane half)
  const int kh = mh << 1;                // K offset 0 or 2 inside the x4 step
  const float* arow = x + (size_t)(row0 + (lane & 15)) * SOM_D + kh;
  const float cterm = (float)SOM_D * SOM_EPS * SOM_EPS;

  for (int ct = 0; ct < 32; ++ct) {
    const int ncol = (wave + (ct << 3)) << 4;   // this wave's 16-col tile
    const int nl   = ncol + (lane & 15);
    const float* bp = w + (size_t)kh * SOM_K + nl;

    v8f acc = {};
    for (int k = 0; k < SOM_D; k += 4) {
      v2f a = *(const v2f*)(arow + k);          // global_load_b64
      v2f b;
      b.x = bp[(size_t)k * SOM_K];
      b.y = bp[(size_t)(k + 1) * SOM_K];
      acc = __builtin_amdgcn_wmma_f32_16x16x4_f32(
          /*neg_a=*/false, a, /*neg_b=*/false, b,
          /*c_mod=*/(short)0, acc, /*reuse_a=*/false, /*reuse_b=*/false);
    }

    const float w2v = w2g[nl];
    const float swv = swg[nl];
#pragma unroll
    for (int r = 0; r < 8; ++r) {
      const int m = r + (mh << 3);
      const float cross = acc[r];
      // ||x - w + eps||^2 = x2 + w2 - 2 x.w + 2 eps (sx - sw) + d eps^2
      float sq = s_x2[m] + w2v - 2.f * cross
               + 2.f * SOM_EPS * (s_sx[m] - swv) + cterm;
      s_dist[m * SOM_K + nl] = sqrtf(fmaxf(sq, 0.f));
    }
  }
  __syncthreads();

  // ---- Stage 2: per-row min / argmin (first-index ties, like jnp.argmin) ----
  {
    const int m = tid >> 4, c0 = tid & 15;
    const float* dr = s_dist + m * SOM_K;
    float mn = 3.4e38f;
    int   mi = 0;
    for (int k = c0; k < SOM_K; k += 16) {
      float v = dr[k];
      if (v < mn) { mn = v; mi = k; }
    }
    s_pmin[tid] = mn;
    s_pidx[tid] = mi;
  }
  __syncthreads();
  if (tid < 16) {
    const int m = tid;
    float mn = s_pmin[m * 16];
    int   mi = s_pidx[m * 16];
    for (int c = 1; c < 16; ++c) {
      float v = s_pmin[m * 16 + c];
      int   i = s_pidx[m * 16 + c];
      if (v < mn || (v == mn && i < mi)) { mn = v; mi = i; }
    }
    s_rowmin[m] = mn;
    float bi = (float)(mi >> 6);    // locations = meshgrid(ij), 64x64
    float bj = (float)(mi & 63);
    s_bi[m] = bi;
    s_bj[m] = bj;
    bmu_out[(size_t)(row0 + m) * 2 + 0] = bi;
    bmu_out[(size_t)(row0 + m) * 2 + 1] = bj;
  }
  __syncthreads();
  if (tid == 0) {
    float s = 0.f;
    for (int m = 0; m < 16; ++m) s += s_rowmin[m];
    atomicAdd(loss_out, s * (1.0f / (float)SOM_N));
  }

  // ---- Stage 3: som_output = exp(loss - dist) * gaussian(grid dist^2) ----
  for (int m = 0; m < 16; ++m) {
    const float bi = s_bi[m], bj = s_bj[m], lm = s_rowmin[m];
    const float* dr = s_dist + m * SOM_K;
    float* orow = som_out + (size_t)(row0 + m) * SOM_K;
    for (int k = tid * 4; k < SOM_K; k += 1024) {
      float4 d4 = *(const float4*)(dr + k);     // ds_load_b128
      const float dd[4] = {d4.x, d4.y, d4.z, d4.w};
#pragma unroll
      for (int u = 0; u < 4; ++u) {
        const int kk = k + u;
        float di = (float)(kk >> 6) - bi;
        float dj = (float)(kk & 63) - bj;
        float lr = __expf(-(di * di + dj * dj) * SOM_INV_SIG2);
        orow[kk] = __expf(lm - dd[u]) * lr;     // out base only 4B-aligned
      }
    }
  }
}

// ------------------------------------------------------------------ launch
extern "C" void kernel_launch(void* const* d_in, const int* in_sizes, int n_in,
                              void* d_out, int out_size, void* d_ws, size_t ws_size,
                              hipStream_t stream) {
  const float* x = (const float*)d_in[0];   // (16384, 1024)
  const float* w = (const float*)d_in[1];   // (1024, 4096)
  // d_in[2] (locations) is an implicit 64x64 meshgrid; recomputed on the fly.

  float* x2 = (float*)d_ws;                 // 16384
  float* sx = x2 + SOM_N;                   // 16384
  float* w2 = sx + SOM_N;                   // 4096
  float* sw = w2 + SOM_K;                   // 4096  (160 KB total)

  float* bmu  = (float*)d_out;              // (N,1,2) -> 32768 floats
  float* loss = bmu + 2 * SOM_N;            // 1 float
  float* som  = loss + 1;                   // (N,K)

  som_row_stats<<<SOM_N / 8, 256, 0, stream>>>(x, x2, sx);
  som_col_stats<<<SOM_K / 256, 256, 0, stream>>>(w, w2, sw);
  som_zero<<<1, 1, 0, stream>>>(loss);
  som_main<<<SOM_N / 16, 256, 0, stream>>>(x, w, x2, sx, w2, sw, bmu, loss, som);
}